// GeometricAttention_25683904430751
// MI455X (gfx1250) — compile-verified
//
#include <hip/hip_runtime.h>
#include <math.h>

typedef __attribute__((ext_vector_type(2))) float v2f;
typedef __attribute__((ext_vector_type(8))) float v8f;

#define S_TOK 2048
#define DIM   1024
#define NH    16
#define PROJ  240      // (4+M)*H*3
#define NROT  144      // (2+M)*H*3
#define COUT  48       // H*3*M
#define NEG_MIN (-3.402823466e38f)
#define INV_S3  (0.57735026918962576f)

// ---------------------------------------------------------------------------
// K1: LayerNorm  nx = (x - mu) / sqrt(var + 1e-5) * ln_w
// ---------------------------------------------------------------------------
__global__ void ln_kernel(const float* __restrict__ x,
                          const float* __restrict__ w,
                          float* __restrict__ nx) {
  __shared__ float ssum[8], ssq[8];
  const int s = blockIdx.x;
  const float* xr = x + (size_t)s * DIM;
  float lsum = 0.f, lsq = 0.f;
  for (int d = threadIdx.x; d < DIM; d += 256) {
    float v = xr[d];
    lsum += v; lsq += v * v;
  }
#pragma unroll
  for (int msk = 16; msk >= 1; msk >>= 1) {
    lsum += __shfl_xor(lsum, msk, 32);
    lsq  += __shfl_xor(lsq,  msk, 32);
  }
  const int wid = threadIdx.x >> 5;
  if ((threadIdx.x & 31) == 0) { ssum[wid] = lsum; ssq[wid] = lsq; }
  __syncthreads();
  float tsum = 0.f, tsq = 0.f;
#pragma unroll
  for (int i = 0; i < 8; ++i) { tsum += ssum[i]; tsq += ssq[i]; }
  const float mu   = tsum * (1.0f / DIM);
  const float var  = tsq * (1.0f / DIM) - mu * mu;
  const float rstd = rsqrtf(var + 1e-5f);
  float* nr = nx + (size_t)s * DIM;
  for (int d = threadIdx.x; d < DIM; d += 256)
    nr[d] = (xr[d] - mu) * rstd * w[d];
}

// ---------------------------------------------------------------------------
// K2/K6: C[M][N] = A[M][K] @ B[N][K]^T using chained V_WMMA_F32_16X16X4_F32.
// One wave = one 16x16 C tile. K must be a multiple of 4.
// A lane layout (16x4):  lane l -> row l&15, K pair = 2*(l>>4)
// B lane layout (4x16):  lane l -> col l&15, K pair = 2*(l>>4)
// C lane layout (16x16): lane l, vgpr v -> row 8*(l>>4)+v, col l&15
// ---------------------------------------------------------------------------
__global__ void gemm_wmma(const float* __restrict__ A,
                          const float* __restrict__ B,
                          float* __restrict__ C,
                          int K, int lda, int ldb, int ldc) {
  const int l   = threadIdx.x;
  const int ln  = l & 15;
  const int hi  = l >> 4;
  const int kh  = hi * 2;
  const float* ar = A + (size_t)(blockIdx.x * 16 + ln) * lda + kh;
  const float* br = B + (size_t)(blockIdx.y * 16 + ln) * ldb + kh;
  v8f c = {};
  for (int k0 = 0; k0 < K; k0 += 4) {
    v2f a = *(const v2f*)(ar + k0);
    v2f b = *(const v2f*)(br + k0);
    c = __builtin_amdgcn_wmma_f32_16x16x4_f32(false, a, false, b,
                                              (short)0, c, false, false);
  }
  const int crow = blockIdx.x * 16 + 8 * hi;
  const int ccol = blockIdx.y * 16 + ln;
#pragma unroll
  for (int v = 0; v < 8; ++v)
    C[(size_t)(crow + v) * ldc + ccol] = c[v];
}

// ---------------------------------------------------------------------------
// K3: per-(token,head) rotation of projected vectors into attention operands.
// Layouts: q_rot/k_rot/q_dist/k_dist/val : [H][S][4] (pad 4th = 0)
//          q2/k2 : [H][S]
// ---------------------------------------------------------------------------
__global__ void qkv_kernel(const float* __restrict__ p,
                           const float* __restrict__ rot,
                           const float* __restrict__ trans,
                           float* __restrict__ q_rot, float* __restrict__ k_rot,
                           float* __restrict__ valv,
                           float* __restrict__ q_dist, float* __restrict__ k_dist,
                           float* __restrict__ q2, float* __restrict__ k2) {
  const int t = blockIdx.x * blockDim.x + threadIdx.x;
  if (t >= S_TOK * NH) return;
  const int s = t >> 4;
  const int h = t & 15;
  const float* R  = rot + (size_t)s * 9;      // R[e*3+d]
  const float* pr = p + (size_t)s * PROJ;
  const float* tr = trans + (size_t)s * 3;

  float qr[3], kr[3], vv[3], qd[3], kd[3];
#pragma unroll
  for (int e = 0; e < 3; ++e) {
    const float r0 = R[e * 3 + 0], r1 = R[e * 3 + 1], r2 = R[e * 3 + 2];
    const float* a;
    a = pr + h * 3;               qr[e] = r0 * a[0] + r1 * a[1] + r2 * a[2];
    a = pr + (16 + h) * 3;        kr[e] = r0 * a[0] + r1 * a[1] + r2 * a[2];
    a = pr + (32 + h) * 3;        vv[e] = r0 * a[0] + r1 * a[1] + r2 * a[2];
    a = pr + NROT + h * 3;        qd[e] = r0 * a[0] + r1 * a[1] + r2 * a[2] + tr[e];
    a = pr + NROT + (16 + h) * 3; kd[e] = r0 * a[0] + r1 * a[1] + r2 * a[2] + tr[e];
  }
  const size_t o = ((size_t)h * S_TOK + s) * 4;
#pragma unroll
  for (int e = 0; e < 3; ++e) {
    q_rot[o + e]  = qr[e];
    k_rot[o + e]  = kr[e];
    valv[o + e]   = vv[e];
    q_dist[o + e] = qd[e];
    k_dist[o + e] = kd[e];
  }
  q_rot[o + 3] = 0.f; k_rot[o + 3] = 0.f; valv[o + 3] = 0.f;
  q_dist[o + 3] = 0.f; k_dist[o + 3] = 0.f;
  q2[(size_t)h * S_TOK + s] = qd[0] * qd[0] + qd[1] * qd[1] + qd[2] * qd[2];
  k2[(size_t)h * S_TOK + s] = kd[0] * kd[0] + kd[1] * kd[1] + kd[2] * kd[2];
}

// ---------------------------------------------------------------------------
// K4: streaming geometric attention.
// grid = (H, S/16), block = 32 (one wave per 16-query tile per head).
// Per 16-key tile: 2x V_WMMA_F32_16X16X4_F32 (rot scores, q.k for distance),
// each lane handles one key column of the C tile with online softmax and a
// per-lane 3-dim value accumulator; split-softmax combine via shfl_xor at end.
// Output o1: [S][COUT] = (s, h*3+d).
// ---------------------------------------------------------------------------
__global__ void attn_kernel(const float* __restrict__ q_rot,
                            const float* __restrict__ k_rot,
                            const float* __restrict__ q_dist,
                            const float* __restrict__ k_dist,
                            const float* __restrict__ valv,
                            const float* __restrict__ q2w,
                            const float* __restrict__ k2w,
                            const int* __restrict__ seq_id,
                            const unsigned char* __restrict__ amask,
                            const float* __restrict__ dist_scale,
                            const float* __restrict__ rot_scale,
                            float* __restrict__ o1) {
  const int l  = threadIdx.x;
  const int ln = l & 15;
  const int hi = l >> 4;
  const int kh = hi * 2;
  const int h  = blockIdx.x;
  const int qt = blockIdx.y;

  const float rw = log1pf(__expf(rot_scale[h]))  * INV_S3;
  const float dw = log1pf(__expf(dist_scale[h])) * INV_S3;

  // A operands: 16 query rows x 4 (K pad)
  const int arow = qt * 16 + ln;
  const float* qrb = q_rot  + ((size_t)h * S_TOK + arow) * 4 + kh;
  const float* qdb = q_dist + ((size_t)h * S_TOK + arow) * 4 + kh;
  const v2f a_rot  = *(const v2f*)qrb;
  const v2f a_dist = *(const v2f*)qdb;

  // Per-lane C rows
  const int rbase = qt * 16 + 8 * hi;
  float q2r[8]; int seqr[8];
#pragma unroll
  for (int v = 0; v < 8; ++v) {
    q2r[v]  = q2w[(size_t)h * S_TOK + rbase + v];
    seqr[v] = seq_id[rbase + v];
  }

  float m[8], lsum[8], a0[8], a1[8], a2[8];
#pragma unroll
  for (int v = 0; v < 8; ++v) {
    m[v] = NEG_MIN; lsum[v] = 0.f; a0[v] = 0.f; a1[v] = 0.f; a2[v] = 0.f;
  }

  for (int kt = 0; kt < S_TOK / 16; ++kt) {
    const int bcol = kt * 16 + ln;
    const v2f b_rot  = *(const v2f*)(k_rot  + ((size_t)h * S_TOK + bcol) * 4 + kh);
    const v2f b_dist = *(const v2f*)(k_dist + ((size_t)h * S_TOK + bcol) * 4 + kh);
    const v8f zc = {};
    v8f c_rot = __builtin_amdgcn_wmma_f32_16x16x4_f32(false, a_rot,  false, b_rot,
                                                      (short)0, zc, false, false);
    v8f c_qk  = __builtin_amdgcn_wmma_f32_16x16x4_f32(false, a_dist, false, b_dist,
                                                      (short)0, zc, false, false);
    const float k2c  = k2w[(size_t)h * S_TOK + bcol];
    const int   seqc = seq_id[bcol];
    const bool  mok  = amask[bcol] != 0;
    const float* vb  = valv + ((size_t)h * S_TOK + bcol) * 4;
    const float v0 = vb[0], v1 = vb[1], v2c = vb[2];
#pragma unroll
    for (int r = 0; r < 8; ++r) {
      const float sq   = q2r[r] + k2c - 2.0f * c_qk[r];
      const float dist = sqrtf(fmaxf(sq, 0.0f));
      const float bias = mok ? ((seqr[r] == seqc) ? 1.0f : 0.0f) : NEG_MIN;
      const float sc   = c_rot[r] * rw - dist * dw + bias;
      const float nm   = fmaxf(m[r], sc);
      const float f    = __expf(m[r] - nm);
      const float e    = __expf(sc - nm);
      m[r]    = nm;
      lsum[r] = lsum[r] * f + e;
      a0[r]   = a0[r] * f + e * v0;
      a1[r]   = a1[r] * f + e * v1;
      a2[r]   = a2[r] * f + e * v2c;
    }
  }

  // split-softmax combine across the 16 lanes sharing the same rows
#pragma unroll
  for (int r = 0; r < 8; ++r) {
    float mm = m[r], ll = lsum[r], x0 = a0[r], x1 = a1[r], x2 = a2[r];
#pragma unroll
    for (int msk = 8; msk >= 1; msk >>= 1) {
      const float mo = __shfl_xor(mm, msk, 32);
      const float lo = __shfl_xor(ll, msk, 32);
      const float y0 = __shfl_xor(x0, msk, 32);
      const float y1 = __shfl_xor(x1, msk, 32);
      const float y2 = __shfl_xor(x2, msk, 32);
      const float nm = fmaxf(mm, mo);
      const float f1 = __expf(mm - nm);
      const float f2 = __expf(mo - nm);
      ll = ll * f1 + lo * f2;
      x0 = x0 * f1 + y0 * f2;
      x1 = x1 * f1 + y1 * f2;
      x2 = x2 * f1 + y2 * f2;
      mm = nm;
    }
    if (ln == 0) {
      const int row = rbase + r;
      const float invl = 1.0f / ll;
      float* op = o1 + (size_t)row * COUT + h * 3;
      op[0] = x0 * invl; op[1] = x1 * invl; op[2] = x2 * invl;
    }
  }
}

// ---------------------------------------------------------------------------
// K5: rotate attention output back (out[n,e] = sum_d o[n,d] * rot[d,e]) and
// apply affine mask over the query dimension.
// ---------------------------------------------------------------------------
__global__ void rotout_kernel(const float* __restrict__ o1,
                              const float* __restrict__ rot,
                              const unsigned char* __restrict__ amask,
                              float* __restrict__ rout) {
  const int t = blockIdx.x * blockDim.x + threadIdx.x;
  if (t >= S_TOK * NH) return;
  const int s = t >> 4;
  const int h = t & 15;
  const float* R  = rot + (size_t)s * 9;
  const float* ip = o1  + (size_t)s * COUT + h * 3;
  float* op       = rout + (size_t)s * COUT + h * 3;
  const float msk = amask[s] ? 1.0f : 0.0f;
  const float d0 = ip[0], d1 = ip[1], d2 = ip[2];
#pragma unroll
  for (int e = 0; e < 3; ++e)
    op[e] = msk * (d0 * R[0 * 3 + e] + d1 * R[1 * 3 + e] + d2 * R[2 * 3 + e]);
}

// ---------------------------------------------------------------------------
extern "C" void kernel_launch(void* const* d_in, const int* in_sizes, int n_in,
                              void* d_out, int out_size, void* d_ws, size_t ws_size,
                              hipStream_t stream) {
  const float* x      = (const float*)d_in[0];
  const float* trans  = (const float*)d_in[1];
  const float* rot    = (const float*)d_in[2];
  const unsigned char* amask = (const unsigned char*)d_in[3];
  const int*   seq    = (const int*)d_in[4];
  const float* ln_w   = (const float*)d_in[5];
  const float* W_proj = (const float*)d_in[6];   // [PROJ][DIM]
  const float* W_out  = (const float*)d_in[7];   // [DIM][COUT]
  const float* dsc    = (const float*)d_in[8];
  const float* rsc    = (const float*)d_in[9];
  float* out = (float*)d_out;

  float* ws = (float*)d_ws;
  size_t off = 0;
  float* nx     = ws + off; off += (size_t)S_TOK * DIM;     // 2M
  float* p      = ws + off; off += (size_t)S_TOK * PROJ;    // 480K
  float* q_rot  = ws + off; off += (size_t)NH * S_TOK * 4;
  float* k_rot  = ws + off; off += (size_t)NH * S_TOK * 4;
  float* q_dist = ws + off; off += (size_t)NH * S_TOK * 4;
  float* k_dist = ws + off; off += (size_t)NH * S_TOK * 4;
  float* valv   = ws + off; off += (size_t)NH * S_TOK * 4;
  float* q2     = ws + off; off += (size_t)NH * S_TOK;
  float* k2     = ws + off; off += (size_t)NH * S_TOK;
  float* o1     = ws + off; off += (size_t)S_TOK * COUT;
  float* rout   = ws + off; off += (size_t)S_TOK * COUT;

  // K1: layernorm
  ln_kernel<<<S_TOK, 256, 0, stream>>>(x, ln_w, nx);

  // K2: p = nx @ W_proj^T   (M=2048, N=240, K=1024)
  {
    dim3 g(S_TOK / 16, PROJ / 16);
    gemm_wmma<<<g, 32, 0, stream>>>(nx, W_proj, p, DIM, DIM, DIM, PROJ);
  }

  // K3: rotate projected vectors into attention operands
  {
    const int total = S_TOK * NH;
    qkv_kernel<<<(total + 255) / 256, 256, 0, stream>>>(
        p, rot, trans, q_rot, k_rot, valv, q_dist, k_dist, q2, k2);
  }

  // K4: streaming attention
  {
    dim3 g(NH, S_TOK / 16);
    attn_kernel<<<g, 32, 0, stream>>>(q_rot, k_rot, q_dist, k_dist, valv,
                                      q2, k2, seq, amask, dsc, rsc, o1);
  }

  // K5: rotate back + query mask
  {
    const int total = S_TOK * NH;
    rotout_kernel<<<(total + 255) / 256, 256, 0, stream>>>(o1, rot, amask, rout);
  }

  // K6: out = rout @ W_out^T  (M=2048, N=1024, K=48)
  {
    dim3 g(S_TOK / 16, DIM / 16);
    gemm_wmma<<<g, 32, 0, stream>>>(rout, W_out, out, COUT, COUT, COUT, DIM);
  }
}